// SpatialGAT_6511170421714
// MI455X (gfx1250) — compile-verified
//
#include <hip/hip_runtime.h>
#include <math.h>

// ---------------------------------------------------------------------------
// SpatialGAT for MI455X (gfx1250, wave32).
// Dense projections use V_WMMA_F32_16X16X4_F32 with K padded in memory
// (xin stride 24, W1 zero-padded to 24x64) so the inner loop has zero
// bounds-check branches: one global_load_b64 A-frag + 8 b32 B-frags -> 4 WMMA.
// Edge softmax/aggregation is 3-pass atomics; all intermediate buffers
// (~160MB) fit in the 192MB L2, so random edge traffic is L2-resident.
// ---------------------------------------------------------------------------

typedef __attribute__((ext_vector_type(2))) float v2f;
typedef __attribute__((ext_vector_type(8))) float v8f;

static __device__ __forceinline__ void atomicMaxF(float* addr, float val) {
  // sign-split trick: positive floats order as signed ints,
  // negative floats order reversed as unsigned ints.
  if (val >= 0.0f) {
    atomicMax((int*)addr, __float_as_int(val));
  } else {
    atomicMin((unsigned int*)addr, __float_as_uint(val));
  }
}

__global__ void fill_f32(float* __restrict__ p, float v, int n) {
  int i = blockIdx.x * blockDim.x + threadIdx.x;
  if (i < n) p[i] = v;
}

// xin[i, 0:5] = x[i,:], xin[i, 5:21] = type_emb[type_ids[i],:], xin[i,21:24]=0
__global__ void build_xin(const float* __restrict__ x, const int* __restrict__ tids,
                          const float* __restrict__ temb, float* __restrict__ xin, int n) {
  int i = blockIdx.x * blockDim.x + threadIdx.x;
  if (i >= n) return;
  float* o = xin + (size_t)i * 24;
#pragma unroll
  for (int j = 0; j < 5; ++j) o[j] = x[(size_t)i * 5 + j];
  const float* e = temb + (size_t)tids[i] * 16;
#pragma unroll
  for (int j = 0; j < 16; ++j) o[5 + j] = e[j];
  o[21] = 0.0f; o[22] = 0.0f; o[23] = 0.0f;
}

// W1p[k,c] = W1[k,c] for k<21, else 0.  (24 x 64)
__global__ void pad_w1(const float* __restrict__ W1, float* __restrict__ W1p) {
  int i = blockIdx.x * blockDim.x + threadIdx.x;  // 0..1535
  if (i >= 24 * 64) return;
  const int k = i >> 6;
  W1p[i] = (k < 21) ? W1[i] : 0.0f;
}

// 4 waves per block; each wave computes one 16x64 tile of C = A[.,K] @ B[K,64]
// via V_WMMA_F32_16X16X4_F32.  K must be a multiple of 4 (pre-padded).
// A-frag: v2f per lane (lanes 16-31 carry K+2).  B-frag: 2 VGPRs, N over
// lane&15, upper lanes carry K+2.  C/D: 8 VGPRs, M = i (+8 for upper lanes).
template <int K>
__global__ void gemm16x64_wmma(const float* __restrict__ A, const float* __restrict__ B,
                               float* __restrict__ C, int nrows) {
  const int lane = threadIdx.x & 31;
  const int wave = threadIdx.x >> 5;
  const int ml = lane & 15;
  const int kh = (lane >> 4) << 1;  // 0 or 2
  const int row0 = (blockIdx.x * 4 + wave) * 16;
  if (row0 >= nrows) return;
  const int r = row0 + ml;
  const int rl = (r < nrows) ? r : (nrows - 1);  // clamp loads; stores guarded

  v8f acc0 = {}, acc1 = {}, acc2 = {}, acc3 = {};
  const float* Arow = A + (size_t)rl * K;
#pragma unroll
  for (int kk = 0; kk < K; kk += 4) {
    const int ka = kk + kh;
    const v2f a = *(const v2f*)(Arow + ka);  // 8B-aligned: K, kh, kk all even

    const float* Br0 = B + (size_t)ka * 64;
    const float* Br1 = Br0 + 64;
    v2f b0, b1, b2, b3;
    b0.x = Br0[ml +  0];  b0.y = Br1[ml +  0];
    b1.x = Br0[ml + 16];  b1.y = Br1[ml + 16];
    b2.x = Br0[ml + 32];  b2.y = Br1[ml + 32];
    b3.x = Br0[ml + 48];  b3.y = Br1[ml + 48];

    acc0 = __builtin_amdgcn_wmma_f32_16x16x4_f32(false, a, false, b0, (short)0, acc0, false, false);
    acc1 = __builtin_amdgcn_wmma_f32_16x16x4_f32(false, a, false, b1, (short)0, acc1, false, false);
    acc2 = __builtin_amdgcn_wmma_f32_16x16x4_f32(false, a, false, b2, (short)0, acc2, false, false);
    acc3 = __builtin_amdgcn_wmma_f32_16x16x4_f32(false, a, false, b3, (short)0, acc3, false, false);
  }

  const int rbase = row0 + ((lane >> 4) << 3);
#pragma unroll
  for (int i = 0; i < 8; ++i) {
    const int rr = rbase + i;
    if (rr < nrows) {
      float* Cr = C + (size_t)rr * 64;
      Cr[ml +  0] = acc0[i];
      Cr[ml + 16] = acc1[i];
      Cr[ml + 32] = acc2[i];
      Cr[ml + 48] = acc3[i];
    }
  }
}

// als[n,h] = <h[n,h,:], a_src[h,:]>, ald likewise.  heads*ch == 64.
__global__ void attn_logits(const float* __restrict__ h, const float* __restrict__ asrc,
                            const float* __restrict__ adst, float* __restrict__ als,
                            float* __restrict__ ald, int n, int heads, int ch) {
  int i = blockIdx.x * blockDim.x + threadIdx.x;
  if (i >= n * heads) return;
  const int node = i / heads, hd = i - node * heads;
  const float* hp = h + (size_t)node * 64 + hd * ch;
  const float* as = asrc + hd * ch;
  const float* ad = adst + hd * ch;
  float s0 = 0.0f, s1 = 0.0f;
  for (int c = 0; c < ch; ++c) {
    const float v = hp[c];
    s0 += v * as[c];
    s1 += v * ad[c];
  }
  als[i] = s0;
  ald[i] = s1;
}

static __device__ __forceinline__ float leaky(float v) {
  return (v > 0.0f) ? v : 0.2f * v;
}

__global__ void edge_max(const int* __restrict__ src, const int* __restrict__ dst,
                         const float* __restrict__ als, const float* __restrict__ ald,
                         float* __restrict__ m, int E, int Etot, int H) {
  int e = blockIdx.x * blockDim.x + threadIdx.x;
  if (e >= Etot) return;
  const int s = (e < E) ? src[e] : (e - E);
  const int d = (e < E) ? dst[e] : (e - E);
  for (int h = 0; h < H; ++h) {
    const float v = leaky(als[s * H + h] + ald[d * H + h]);
    atomicMaxF(&m[d * H + h], v);
  }
}

__global__ void edge_expsum(const int* __restrict__ src, const int* __restrict__ dst,
                            const float* __restrict__ als, const float* __restrict__ ald,
                            const float* __restrict__ m, float* __restrict__ ssum,
                            int E, int Etot, int H) {
  int e = blockIdx.x * blockDim.x + threadIdx.x;
  if (e >= Etot) return;
  const int s = (e < E) ? src[e] : (e - E);
  const int d = (e < E) ? dst[e] : (e - E);
  for (int h = 0; h < H; ++h) {
    const float v = leaky(als[s * H + h] + ald[d * H + h]);
    atomicAdd(&ssum[d * H + h], __expf(v - m[d * H + h]));
  }
}

// One 32-lane wave per edge; lane handles channels c and c+32.
__global__ void edge_aggregate(const int* __restrict__ src, const int* __restrict__ dst,
                               const float* __restrict__ als, const float* __restrict__ ald,
                               const float* __restrict__ m, const float* __restrict__ ssum,
                               const float* __restrict__ h, float* __restrict__ agg,
                               int E, int Etot, int H) {
  const int g = blockIdx.x * (blockDim.x >> 5) + (threadIdx.x >> 5);
  const int lane = threadIdx.x & 31;
  if (g >= Etot) return;
  const int s = (g < E) ? src[g] : (g - E);
  const int d = (g < E) ? dst[g] : (g - E);
  float alpha0, alpha1;
  {
    const float v = leaky(als[s * H] + ald[d * H]);
    alpha0 = __expf(v - m[d * H]) / (ssum[d * H] + 1e-16f);
  }
  if (H == 2) {
    const float v = leaky(als[s * H + 1] + ald[d * H + 1]);
    alpha1 = __expf(v - m[d * H + 1]) / (ssum[d * H + 1] + 1e-16f);
  } else {
    alpha1 = alpha0;
  }
  const int c0 = lane, c1 = lane + 32;
  atomicAdd(&agg[(size_t)d * 64 + c0], alpha0 * h[(size_t)s * 64 + c0]);
  atomicAdd(&agg[(size_t)d * 64 + c1], alpha1 * h[(size_t)s * 64 + c1]);
}

__global__ void bias_relu(const float* __restrict__ agg, const float* __restrict__ b,
                          float* __restrict__ out, int total) {
  int i = blockIdx.x * blockDim.x + threadIdx.x;
  if (i >= total) return;
  const float v = agg[i] + b[i & 63];
  out[i] = (v > 0.0f) ? v : 0.0f;
}

// One wave per node: add bias, LayerNorm over 64 channels, scale/shift.
__global__ void layernorm_out(const float* __restrict__ agg, const float* __restrict__ b2,
                              const float* __restrict__ gamma, const float* __restrict__ beta,
                              float* __restrict__ out, int n) {
  const int node = blockIdx.x * (blockDim.x >> 5) + (threadIdx.x >> 5);
  const int lane = threadIdx.x & 31;
  if (node >= n) return;
  const float v0 = agg[(size_t)node * 64 + lane]      + b2[lane];
  const float v1 = agg[(size_t)node * 64 + lane + 32] + b2[lane + 32];
  float s = v0 + v1;
#pragma unroll
  for (int o = 16; o > 0; o >>= 1) s += __shfl_xor(s, o, 32);
  const float mu = s * (1.0f / 64.0f);
  const float d0 = v0 - mu, d1 = v1 - mu;
  float q = d0 * d0 + d1 * d1;
#pragma unroll
  for (int o = 16; o > 0; o >>= 1) q += __shfl_xor(q, o, 32);
  const float inv = rsqrtf(q * (1.0f / 64.0f) + 1e-5f);
  out[(size_t)node * 64 + lane]      = d0 * inv * gamma[lane]      + beta[lane];
  out[(size_t)node * 64 + lane + 32] = d1 * inv * gamma[lane + 32] + beta[lane + 32];
}

static inline int cdiv(long long a, long long b) { return (int)((a + b - 1) / b); }

extern "C" void kernel_launch(void* const* d_in, const int* in_sizes, int n_in,
                              void* d_out, int out_size, void* d_ws, size_t ws_size,
                              hipStream_t stream) {
  const float* x       = (const float*)d_in[0];   // [N,5]
  const int*   eidx    = (const int*)  d_in[1];   // [2,E]
  const int*   tids    = (const int*)  d_in[2];   // [N]
  const float* temb    = (const float*)d_in[3];   // [8,16]
  const float* W1      = (const float*)d_in[4];   // [21,64]
  const float* a_src1  = (const float*)d_in[5];   // [2,32]
  const float* a_dst1  = (const float*)d_in[6];   // [2,32]
  const float* b1      = (const float*)d_in[7];   // [64]
  const float* W2      = (const float*)d_in[8];   // [64,64]
  const float* a_src2  = (const float*)d_in[9];   // [1,64]
  const float* a_dst2  = (const float*)d_in[10];  // [1,64]
  const float* b2      = (const float*)d_in[11];  // [64]
  const float* gamma   = (const float*)d_in[12];  // [64]
  const float* beta    = (const float*)d_in[13];  // [64]

  const int N = in_sizes[0] / 5;
  const int E = in_sizes[1] / 2;
  const int Etot = E + N;  // with self loops
  const int* src = eidx;
  const int* dst = eidx + E;

  float* ws = (float*)d_ws;
  size_t o = 0;
  float* xin   = ws + o; o += (size_t)N * 24;  // padded K=24
  float* W1p   = ws + o; o += (size_t)24 * 64; // zero-padded W1
  float* h1    = ws + o; o += (size_t)N * 64;
  float* als1  = ws + o; o += (size_t)N * 2;
  float* ald1  = ws + o; o += (size_t)N * 2;
  float* m1    = ws + o; o += (size_t)N * 2;
  float* s1    = ws + o; o += (size_t)N * 2;
  float* agg1  = ws + o; o += (size_t)N * 64;
  float* hrelu = ws + o; o += (size_t)N * 64;
  float* h2    = ws + o; o += (size_t)N * 64;
  float* als2  = ws + o; o += (size_t)N;
  float* ald2  = ws + o; o += (size_t)N;
  float* m2    = ws + o; o += (size_t)N;
  float* s2    = ws + o; o += (size_t)N;
  float* agg2  = ws + o; o += (size_t)N * 64;

  const int T = 256;

  // ----- Layer 1 -----
  build_xin<<<cdiv(N, T), T, 0, stream>>>(x, tids, temb, xin, N);
  pad_w1<<<cdiv(24 * 64, T), T, 0, stream>>>(W1, W1p);
  gemm16x64_wmma<24><<<cdiv(N, 64), 128, 0, stream>>>(xin, W1p, h1, N);
  attn_logits<<<cdiv((long long)N * 2, T), T, 0, stream>>>(h1, a_src1, a_dst1, als1, ald1, N, 2, 32);

  fill_f32<<<cdiv((long long)N * 2, T), T, 0, stream>>>(m1, -INFINITY, N * 2);
  fill_f32<<<cdiv((long long)N * 2, T), T, 0, stream>>>(s1, 0.0f, N * 2);
  fill_f32<<<cdiv((long long)N * 64, T), T, 0, stream>>>(agg1, 0.0f, N * 64);

  edge_max    <<<cdiv(Etot, T), T, 0, stream>>>(src, dst, als1, ald1, m1, E, Etot, 2);
  edge_expsum <<<cdiv(Etot, T), T, 0, stream>>>(src, dst, als1, ald1, m1, s1, E, Etot, 2);
  edge_aggregate<<<cdiv(Etot, 8), 256, 0, stream>>>(src, dst, als1, ald1, m1, s1, h1, agg1, E, Etot, 2);

  bias_relu<<<cdiv((long long)N * 64, T), T, 0, stream>>>(agg1, b1, hrelu, N * 64);

  // ----- Layer 2 -----
  gemm16x64_wmma<64><<<cdiv(N, 64), 128, 0, stream>>>(hrelu, W2, h2, N);
  attn_logits<<<cdiv(N, T), T, 0, stream>>>(h2, a_src2, a_dst2, als2, ald2, N, 1, 64);

  fill_f32<<<cdiv(N, T), T, 0, stream>>>(m2, -INFINITY, N);
  fill_f32<<<cdiv(N, T), T, 0, stream>>>(s2, 0.0f, N);
  fill_f32<<<cdiv((long long)N * 64, T), T, 0, stream>>>(agg2, 0.0f, N * 64);

  edge_max    <<<cdiv(Etot, T), T, 0, stream>>>(src, dst, als2, ald2, m2, E, Etot, 1);
  edge_expsum <<<cdiv(Etot, T), T, 0, stream>>>(src, dst, als2, ald2, m2, s2, E, Etot, 1);
  edge_aggregate<<<cdiv(Etot, 8), 256, 0, stream>>>(src, dst, als2, ald2, m2, s2, h2, agg2, E, Etot, 1);

  // ----- bias + LayerNorm -> d_out -----
  layernorm_out<<<cdiv(N, 8), 256, 0, stream>>>(agg2, b2, gamma, beta, (float*)d_out, N);
}